// reinforce_policy_15118284882447
// MI455X (gfx1250) — compile-verified
//
#include <hip/hip_runtime.h>

typedef float v2f __attribute__((ext_vector_type(2)));
typedef float v8f __attribute__((ext_vector_type(8)));

#define NG 256
#define WPAD 48   // LDS row stride: 2*48 % 64 == 32 -> lane-halves hit disjoint banks

__device__ __forceinline__ unsigned fkey(float f){
  unsigned b = __float_as_uint(f);
  return (b & 0x80000000u) ? ~b : (b | 0x80000000u);
}
__device__ __forceinline__ float finv(unsigned k){
  unsigned b = (k & 0x80000000u) ? (k & 0x7fffffffu) : ~k;
  return __uint_as_float(b);
}

__global__ void k_zero(float* p, long long n){
  long long i = (long long)blockIdx.x * blockDim.x + threadIdx.x;
  long long st = (long long)gridDim.x * blockDim.x;
  for (; i < n; i += st) p[i] = 0.0f;
}

__global__ void k_deg(const int* __restrict__ col, float* __restrict__ deg, long long E){
  long long e = (long long)blockIdx.x * blockDim.x + threadIdx.x;
  if (e < E) atomicAdd(&deg[col[e]], 1.0f);
}

__global__ void k_dinv(float* __restrict__ deg, unsigned* __restrict__ nmaxkey, long long N){
  long long i = (long long)blockIdx.x * blockDim.x + threadIdx.x;
  if (i < N) deg[i] = rsqrtf(deg[i] + 1.0f);   // +1 self loop; deg>=1 always
  if (i < NG) nmaxkey[i] = fkey(-3.0e38f);
}

// out[N,OUTC] = in[N,32] @ W[32,OUTC]  via V_WMMA_F32_16X16X4_F32.
// W staged in LDS (zero-padded to 32 cols) -> branch-free B loads.
template<int OUTC>
__global__ __launch_bounds__(256) void k_gemm32(const float* __restrict__ in,
    const float* __restrict__ W, float* __restrict__ out, long long nrows){
  __shared__ float sW[32 * WPAD];
  int tid = threadIdx.x;
  #pragma unroll 2
  for (int t = tid; t < 32 * 32; t += 256){
    int r = t >> 5, c = t & 31;
    sW[r * WPAD + c] = (c < OUTC) ? W[r * OUTC + c] : 0.0f;
  }
  __syncthreads();

  int lane = tid & 31;
  long long wave = (long long)blockIdx.x * 8 + (tid >> 5);
  long long rowbase = wave * 16;
  if (rowbase >= nrows) return;                 // wave-uniform exit: EXEC stays all-1
  int half = lane >> 4;
  int m = lane & 15;
  long long row = rowbase + m;
  bool rok = row < nrows;
  const float* arow = in + (rok ? row : (nrows - 1)) * 32;

  v8f c0 = {}; v8f c1 = {};
  #pragma unroll
  for (int kk = 0; kk < 8; ++kk){
    int kb = kk * 4 + 2 * half;                 // A/B: VGPR j holds K = kb + j
    v2f a = *(const v2f*)(arow + kb);           // one global_load_b64, 8B-aligned
    a[0] = rok ? a[0] : 0.0f;                   // v_cndmask, no branch
    a[1] = rok ? a[1] : 0.0f;
    v2f b0, b1;
    b0[0] = sW[(kb    ) * WPAD + m];
    b0[1] = sW[(kb + 1) * WPAD + m];
    b1[0] = sW[(kb    ) * WPAD + 16 + m];
    b1[1] = sW[(kb + 1) * WPAD + 16 + m];
    c0 = __builtin_amdgcn_wmma_f32_16x16x4_f32(false, a, false, b0, (short)0, c0, false, false);
    c1 = __builtin_amdgcn_wmma_f32_16x16x4_f32(false, a, false, b1, (short)0, c1, false, false);
  }
  #pragma unroll
  for (int v = 0; v < 8; ++v){                  // D: VGPR v = rows v / v+8 across halves
    long long orow = rowbase + v + 8 * half;
    if (orow < nrows){
      out[orow * OUTC + m] = c0[v];
      if (16 + m < OUTC) out[orow * OUTC + 16 + m] = c1[v];
    }
  }
}

// s[col] += dinv[row]*dinv[col] * h[row]   (one thread per edge-feature;
// edge indices are wave-uniform for F=32 -> broadcast loads, contiguous atomics)
template<int F>
__global__ void k_scatter(const float* __restrict__ h, const int* __restrict__ ei,
    const float* __restrict__ dinv, float* __restrict__ s, long long E){
  long long idx = (long long)blockIdx.x * blockDim.x + threadIdx.x;
  long long e = idx / F;
  int f = (int)(idx - e * F);
  if (e >= E) return;
  int r = ei[e];
  int c = ei[E + e];
  float v = h[(long long)r * F + f] * dinv[r] * dinv[c];
  atomicAdd(&s[(long long)c * F + f], v);
}

__global__ void k_fin1(float* __restrict__ s1, const float* __restrict__ h1,
    const float* __restrict__ dinv, const float* __restrict__ b1, long long N){
  long long idx = (long long)blockIdx.x * blockDim.x + threadIdx.x;
  if (idx >= N * 32) return;
  long long i = idx >> 5; int f = (int)(idx & 31);
  float di = dinv[i];
  s1[idx] = fmaxf(s1[idx] + di * di * h1[idx] + b1[f], 0.0f);
}

__global__ void k_fin2(float* __restrict__ s2, const float* __restrict__ h2,
    const float* __restrict__ dinv, const float* __restrict__ b2,
    const int* __restrict__ batch, float* __restrict__ cnt, float* __restrict__ gsum,
    long long N){
  long long idx = (long long)blockIdx.x * blockDim.x + threadIdx.x;
  if (idx >= N * 24) return;
  long long i = idx / 24; int f = (int)(idx - i * 24);
  float di = dinv[i];
  float v = fmaxf(s2[idx] + di * di * h2[idx] + b2[f], 0.0f);
  s2[idx] = v;
  int b = batch[i];
  atomicAdd(&gsum[b * 24 + f], v);
  if (f == 0) atomicAdd(&cnt[b], 1.0f);
}

__global__ __launch_bounds__(256) void k_nraw(const float* __restrict__ h,
    const float* __restrict__ Wn1, const float* __restrict__ bn1,
    const float* __restrict__ Wn2, const float* __restrict__ bn2,
    const int* __restrict__ batch, float* __restrict__ nraw,
    unsigned* __restrict__ nmaxkey, long long N){
  __shared__ float sW1[24 * 16], sb1[16], sW2[16], sb2;
  int tid = threadIdx.x;
  for (int t = tid; t < 24 * 16; t += 256) sW1[t] = Wn1[t];
  if (tid < 16){ sb1[tid] = bn1[tid]; sW2[tid] = Wn2[tid]; }
  if (tid == 0) sb2 = bn2[0];
  __syncthreads();
  long long i = (long long)blockIdx.x * blockDim.x + tid;
  if (i >= N) return;
  float hr[24];
  #pragma unroll
  for (int f = 0; f < 24; ++f) hr[f] = h[i * 24 + f];
  float nr = sb2;
  #pragma unroll 4
  for (int j = 0; j < 16; ++j){
    float acc = sb1[j];
    #pragma unroll
    for (int f = 0; f < 24; ++f) acc += hr[f] * sW1[f * 16 + j];
    nr += fmaxf(acc, 0.0f) * sW2[j];
  }
  nraw[i] = nr;
  atomicMax(&nmaxkey[batch[i]], fkey(nr));
}

__global__ void k_nexp(float* __restrict__ nraw, const unsigned* __restrict__ nmaxkey,
    const int* __restrict__ batch, float* __restrict__ nsum, long long N){
  long long i = (long long)blockIdx.x * blockDim.x + threadIdx.x;
  if (i >= N) return;
  int b = batch[i];
  float e = expf(nraw[i] - finv(nmaxkey[b]));
  nraw[i] = e;
  atomicAdd(&nsum[b], e);
}

__global__ void k_nfinal(const float* __restrict__ ne, const float* __restrict__ nsum,
    const int* __restrict__ batch, const float* __restrict__ h,
    float* __restrict__ bfeat, float* __restrict__ outn, long long N){
  long long i = (long long)blockIdx.x * blockDim.x + threadIdx.x;
  if (i >= N) return;
  int b = batch[i];
  float nv = ne[i] / nsum[b];
  outn[i] = nv;
  #pragma unroll
  for (int f = 0; f < 24; ++f)
    atomicAdd(&bfeat[b * 24 + f], nv * h[i * 24 + f]);
}

__global__ __launch_bounds__(256) void k_t(const float* __restrict__ gsum,
    const float* __restrict__ cnt, const float* __restrict__ Wg, const float* __restrict__ bg,
    const float* __restrict__ Wt, const float* __restrict__ bt, float* __restrict__ out){
  int g = threadIdx.x;
  float c = fmaxf(cnt[g], 1.0f);
  float gp[24];
  for (int f = 0; f < 24; ++f) gp[f] = gsum[g * 24 + f] / c;
  float gg[32];
  for (int j = 0; j < 32; ++j){
    float acc = bg[j];
    for (int f = 0; f < 24; ++f) acc += gp[f] * Wg[f * 32 + j];
    gg[j] = acc;
  }
  float t0 = bt[0], t1 = bt[1];
  for (int k = 0; k < 32; ++k){ t0 += gg[k] * Wt[k * 2]; t1 += gg[k] * Wt[k * 2 + 1]; }
  float mx = fmaxf(t0, t1);
  float e0 = expf(t0 - mx), e1 = expf(t1 - mx);
  float s = e0 + e1;
  out[g * 2] = e0 / s; out[g * 2 + 1] = e1 / s;
}

__global__ __launch_bounds__(256) void k_bb(const float* __restrict__ bfeat,
    const float* __restrict__ Wb1, const float* __restrict__ bb1,
    const float* __restrict__ Wb2, const float* __restrict__ bb2, float* __restrict__ out){
  __shared__ float sl[NG * 3];
  __shared__ float smax[3], ssum[3];
  int g = threadIdx.x;
  float r[16];
  for (int j = 0; j < 16; ++j){
    float acc = bb1[j];
    for (int f = 0; f < 24; ++f) acc += bfeat[g * 24 + f] * Wb1[f * 16 + j];
    r[j] = fmaxf(acc, 0.0f);
  }
  for (int j = 0; j < 3; ++j){
    float acc = bb2[j];
    for (int k = 0; k < 16; ++k) acc += r[k] * Wb2[k * 3 + j];
    sl[g * 3 + j] = acc;
  }
  __syncthreads();
  if (g < 3){
    float m = -3.0e38f;
    for (int q = 0; q < NG; ++q) m = fmaxf(m, sl[q * 3 + g]);
    float s = 0.0f;
    for (int q = 0; q < NG; ++q) s += expf(sl[q * 3 + g] - m);
    smax[g] = m; ssum[g] = s;
  }
  __syncthreads();
  for (int j = 0; j < 3; ++j)
    out[g * 3 + j] = expf(sl[g * 3 + j] - smax[j]) / ssum[j];
}

extern "C" void kernel_launch(void* const* d_in, const int* in_sizes, int n_in,
                              void* d_out, int out_size, void* d_ws, size_t ws_size,
                              hipStream_t stream){
  const float* x   = (const float*)d_in[0];
  const int*   ei  = (const int*)d_in[1];
  const int*   bat = (const int*)d_in[2];
  const float* W1  = (const float*)d_in[3];
  const float* b1  = (const float*)d_in[4];
  const float* W2  = (const float*)d_in[5];
  const float* b2  = (const float*)d_in[6];
  const float* Wg  = (const float*)d_in[7];
  const float* bg  = (const float*)d_in[8];
  const float* Wt  = (const float*)d_in[9];
  const float* bt  = (const float*)d_in[10];
  const float* Wn1 = (const float*)d_in[11];
  const float* bn1 = (const float*)d_in[12];
  const float* Wn2 = (const float*)d_in[13];
  const float* bn2 = (const float*)d_in[14];
  const float* Wb1 = (const float*)d_in[15];
  const float* bb1 = (const float*)d_in[16];
  const float* Wb2 = (const float*)d_in[17];
  const float* bb2 = (const float*)d_in[18];

  const long long N = in_sizes[0] / 32;
  const long long E = in_sizes[1] / 2;

  // workspace layout: accumulators first (zeroed each call), then transients
  float* ws   = (float*)d_ws;
  float* deg  = ws;                 // N   (becomes dinv in place)
  float* s1   = deg + N;            // 32N (conv1 scatter -> a1 in place)
  float* s2   = s1 + 32 * N;        // 24N (conv2 scatter -> h final in place)
  float* cnt  = s2 + 24 * N;        // 256
  float* gsum = cnt + NG;           // 256*24
  float* bfeat= gsum + NG * 24;     // 256*24
  unsigned* nmaxkey = (unsigned*)(bfeat + NG * 24); // 256
  float* nsum = (float*)(nmaxkey + NG);             // 256
  float* h1   = nsum + NG;          // 32N
  float* h2   = h1 + 32 * N;        // 24N
  float* nraw = h2 + 24 * N;        // N (becomes exp in place)
  long long zeroLen = (long long)((nsum + NG) - ws);

  float* outT = (float*)d_out;      // [256,2]
  float* outN = outT + 2 * NG;      // [N,1]
  float* outB = outN + N;           // [256,3]

  dim3 blk(256);
  k_zero<<<4096, blk, 0, stream>>>(ws, zeroLen);
  k_deg <<<(unsigned)((E + 255) / 256), blk, 0, stream>>>(ei + E, deg, E);
  k_dinv<<<(unsigned)((N + 255) / 256), blk, 0, stream>>>(deg, nmaxkey, N);

  long long waves = (N + 15) / 16;
  unsigned gblocks = (unsigned)((waves + 7) / 8);
  k_gemm32<32><<<gblocks, blk, 0, stream>>>(x, W1, h1, N);
  k_scatter<32><<<(unsigned)((E * 32 + 255) / 256), blk, 0, stream>>>(h1, ei, deg, s1, E);
  k_fin1<<<(unsigned)((N * 32 + 255) / 256), blk, 0, stream>>>(s1, h1, deg, b1, N);

  k_gemm32<24><<<gblocks, blk, 0, stream>>>(s1, W2, h2, N);
  k_scatter<24><<<(unsigned)((E * 24 + 255) / 256), blk, 0, stream>>>(h2, ei, deg, s2, E);
  k_fin2<<<(unsigned)((N * 24 + 255) / 256), blk, 0, stream>>>(s2, h2, deg, b2, bat, cnt, gsum, N);

  unsigned nblocks = (unsigned)((N + 255) / 256);
  k_nraw  <<<nblocks, blk, 0, stream>>>(s2, Wn1, bn1, Wn2, bn2, bat, nraw, nmaxkey, N);
  k_nexp  <<<nblocks, blk, 0, stream>>>(nraw, nmaxkey, bat, nsum, N);
  k_nfinal<<<nblocks, blk, 0, stream>>>(nraw, nsum, bat, s2, bfeat, outN, N);

  k_t <<<1, blk, 0, stream>>>(gsum, cnt, Wg, bg, Wt, bt, outT);
  k_bb<<<1, blk, 0, stream>>>(bfeat, Wb1, bb1, Wb2, bb2, outB);
}